// ColorLoss_65017214927337
// MI455X (gfx1250) — compile-verified
//
#include <hip/hip_runtime.h>
#include <math.h>

typedef __attribute__((ext_vector_type(2))) float v2f;
typedef __attribute__((ext_vector_type(8))) float v8f;

#define HW_PLANE (512 * 512)      // H*W = 262144 (fixed by reference setup)
#define NBLOCKS  1024
#define NTHREADS 256

struct Lab { float L, a, b; };

__device__ __forceinline__ float pow7f(float x) {
    float x2 = x * x;
    float x4 = x2 * x2;
    return x4 * x2 * x;
}

__device__ __forceinline__ float srgb_lin(float c) {
    c = fminf(fmaxf(c, 0.0f), 1.0f);
    // ((c+0.055)/1.055)^2.4 via exp2/log2 (v_exp_f32 / v_log_f32)
    float p = exp2f(2.4f * __log2f((c + 0.055f) * (1.0f / 1.055f)));
    return (c <= 0.04045f) ? c * (1.0f / 12.92f) : p;
}

__device__ __forceinline__ Lab rgb_to_lab(float r, float g, float b) {
    float lr = srgb_lin(r), lg = srgb_lin(g), lb = srgb_lin(b);
    float x = fmaf(0.412453f, lr, fmaf(0.357580f, lg, 0.180423f * lb));
    float y = fmaf(0.212671f, lr, fmaf(0.715160f, lg, 0.072169f * lb));
    float z = fmaf(0.019334f, lr, fmaf(0.119193f, lg, 0.950227f * lb));
    float tx = x * (1.0f / 0.95047f);
    float ty = y;
    float tz = z * (1.0f / 1.08883f);
    const float eps = 0.008856f;
    float fx = (tx > eps) ? exp2f((1.0f / 3.0f) * __log2f(tx)) : fmaf(7.787f, tx, 16.0f / 116.0f);
    float fy = (ty > eps) ? exp2f((1.0f / 3.0f) * __log2f(ty)) : fmaf(7.787f, ty, 16.0f / 116.0f);
    float fz = (tz > eps) ? exp2f((1.0f / 3.0f) * __log2f(tz)) : fmaf(7.787f, tz, 16.0f / 116.0f);
    Lab o;
    o.L = fmaf(116.0f, fy, -16.0f);
    o.a = 500.0f * (fx - fy);
    o.b = 200.0f * (fy - fz);
    return o;
}

// lab1 = labels (true), lab2 = outputs (pred); kL=kC=kH=1
__device__ __forceinline__ float dE00(float ro, float go, float bo,
                                      float rl, float gl, float bl) {
    Lab t = rgb_to_lab(rl, gl, bl);   // lab1
    Lab p = rgb_to_lab(ro, go, bo);   // lab2
    float L1 = t.L, a1 = t.a, b1 = t.b;
    float L2 = p.L, a2 = p.a, b2 = p.b;

    const float P25_7 = 6103515625.0f;   // 25^7
    const float R2D   = 57.29577951308232f;
    const float D2R   = 0.017453292519943295f;

    float C1   = sqrtf(fmaf(a1, a1, b1 * b1));
    float C2   = sqrtf(fmaf(a2, a2, b2 * b2));
    float Cbar = 0.5f * (C1 + C2);
    float c7   = pow7f(Cbar);
    float G    = 0.5f * (1.0f - sqrtf(c7 / (c7 + P25_7)));
    float a1p  = a1 * (1.0f + G);
    float a2p  = a2 * (1.0f + G);
    float C1p  = sqrtf(fmaf(a1p, a1p, b1 * b1));
    float C2p  = sqrtf(fmaf(a2p, a2p, b2 * b2));

    float h1 = R2D * atan2f(b1, a1p); h1 = (h1 < 0.0f) ? h1 + 360.0f : h1;
    float h2 = R2D * atan2f(b2, a2p); h2 = (h2 < 0.0f) ? h2 + 360.0f : h2;

    float dL = L2 - L1;
    float dC = C2p - C1p;
    float dh = h2 - h1;
    dh = (dh >  180.0f) ? dh - 360.0f : dh;
    dh = (dh < -180.0f) ? dh + 360.0f : dh;
    float prodC = C1p * C2p;
    dh = (prodC == 0.0f) ? 0.0f : dh;
    float dH = 2.0f * sqrtf(prodC) * __sinf(D2R * dh * 0.5f);

    float Lbar  = 0.5f * (L1 + L2);
    float Cbarp = 0.5f * (C1p + C2p);
    float hsum  = h1 + h2;
    float habs  = fabsf(h1 - h2);
    float hbar  = (habs <= 180.0f) ? 0.5f * hsum
                : ((hsum < 360.0f) ? 0.5f * (hsum + 360.0f)
                                   : 0.5f * (hsum - 360.0f));
    hbar = (prodC == 0.0f) ? hsum : hbar;

    float T = 1.0f
            - 0.17f * __cosf(D2R * (hbar - 30.0f))
            + 0.24f * __cosf(D2R * (2.0f * hbar))
            + 0.32f * __cosf(D2R * fmaf(3.0f, hbar, 6.0f))
            - 0.20f * __cosf(D2R * fmaf(4.0f, hbar, -63.0f));

    float dt     = (hbar - 275.0f) * (1.0f / 25.0f);
    float dtheta = 30.0f * __expf(-dt * dt);

    float cb7 = pow7f(Cbarp);
    float Rc  = 2.0f * sqrtf(cb7 / (cb7 + P25_7));

    float Lm50   = Lbar - 50.0f;
    float Lm50sq = Lm50 * Lm50;
    float SL = 1.0f + 0.015f * Lm50sq / sqrtf(20.0f + Lm50sq);
    float SC = fmaf(0.045f, Cbarp, 1.0f);
    float SH = fmaf(0.015f * Cbarp, T, 1.0f);
    float RT = -__sinf(D2R * 2.0f * dtheta) * Rc;

    float tL = dL / SL;
    float tC = dC / SC;
    float tH = dH / SH;
    float v  = fmaf(tL, tL, fmaf(tC, tC, fmaf(tH, tH, RT * tC * tH)));
    return sqrtf(fmaxf(v, 0.0f));
}

// Pass 1: per-block partial sums of deltaE over pixel quads (float4 per plane).
__global__ void __launch_bounds__(NTHREADS)
color_loss_partial(const float* __restrict__ outp,
                   const float* __restrict__ labl,
                   float* __restrict__ partials,
                   int nQuads) {
    int tid    = blockIdx.x * blockDim.x + threadIdx.x;
    int stride = gridDim.x * blockDim.x;
    float acc = 0.0f;

    for (int q = tid; q < nQuads; q += stride) {
        int p     = q << 2;                       // pixel index (quad-aligned)
        int batch = p >> 18;                      // / HW_PLANE
        int hw    = p & (HW_PLANE - 1);
        size_t base = (size_t)batch * (3 * HW_PLANE) + (size_t)hw;

        float4 ro = *reinterpret_cast<const float4*>(outp + base);
        float4 go = *reinterpret_cast<const float4*>(outp + base + HW_PLANE);
        float4 bo = *reinterpret_cast<const float4*>(outp + base + 2 * HW_PLANE);
        float4 rl = *reinterpret_cast<const float4*>(labl + base);
        float4 gl = *reinterpret_cast<const float4*>(labl + base + HW_PLANE);
        float4 bl = *reinterpret_cast<const float4*>(labl + base + 2 * HW_PLANE);

        acc += dE00(ro.x, go.x, bo.x, rl.x, gl.x, bl.x);
        acc += dE00(ro.y, go.y, bo.y, rl.y, gl.y, bl.y);
        acc += dE00(ro.z, go.z, bo.z, rl.z, gl.z, bl.z);
        acc += dE00(ro.w, go.w, bo.w, rl.w, gl.w, bl.w);
    }

    __shared__ float red[NTHREADS];
    red[threadIdx.x] = acc;
    __syncthreads();
    for (int s = NTHREADS / 2; s > 0; s >>= 1) {
        if (threadIdx.x < s) red[threadIdx.x] += red[threadIdx.x + s];
        __syncthreads();
    }
    if (threadIdx.x == 0) partials[blockIdx.x] = red[0];
}

// Pass 2: single wave32. Fold up to 1024 partials with chained
// V_WMMA_F32_16X16X4_F32: D = A(data 16x4) x B(ones 4x16) + C.
// With B == all-ones, every D column holds the row-sums of all A's fed so far,
// so total = (sum of this lane's 8 C VGPRs) + (same from lane^16).
__global__ void color_loss_finalize(const float* __restrict__ partials,
                                    float* __restrict__ out,
                                    int nPart, float invN) {
    int lane = threadIdx.x;   // 0..31, EXEC all ones (required for WMMA)
    v8f c = {0.0f, 0.0f, 0.0f, 0.0f, 0.0f, 0.0f, 0.0f, 0.0f};
    v2f ones; ones.x = 1.0f; ones.y = 1.0f;

    for (int i = 0; i < 16; ++i) {          // 16 * 64 = 1024 values
        int idx = i * 64 + lane * 2;
        v2f a;
        a.x = (idx     < nPart) ? partials[idx]     : 0.0f;
        a.y = (idx + 1 < nPart) ? partials[idx + 1] : 0.0f;
        c = __builtin_amdgcn_wmma_f32_16x16x4_f32(
                /*neg_a=*/false, a, /*neg_b=*/false, ones,
                /*c_mod=*/(short)0, c, /*reuse_a=*/false, /*reuse_b=*/false);
    }

    float s = c[0] + c[1] + c[2] + c[3] + c[4] + c[5] + c[6] + c[7];
    s += __shfl_xor(s, 16, 32);             // combine M=0..7 with M=8..15 rows
    if (lane == 0) out[0] = s * invN;
}

extern "C" void kernel_launch(void* const* d_in, const int* in_sizes, int n_in,
                              void* d_out, int out_size, void* d_ws, size_t ws_size,
                              hipStream_t stream) {
    (void)n_in; (void)out_size;
    const float* outputs = (const float*)d_in[0];
    const float* labels  = (const float*)d_in[1];
    float* out      = (float*)d_out;
    float* partials = (float*)d_ws;

    int n       = in_sizes[0];        // 16*3*512*512 = 12582912
    int pixels  = n / 3;              // 4194304
    int nQuads  = pixels / 4;         // 1048576

    int nBlocks = NBLOCKS;
    if ((size_t)nBlocks * sizeof(float) > ws_size)
        nBlocks = (int)(ws_size / sizeof(float));
    if (nBlocks < 1) nBlocks = 1;

    color_loss_partial<<<nBlocks, NTHREADS, 0, stream>>>(outputs, labels, partials, nQuads);

    float invN = 1.0f / (float)pixels;
    color_loss_finalize<<<1, 32, 0, stream>>>(partials, out, nBlocks, invN);
}